// RbfNet_72507637891812
// MI455X (gfx1250) — compile-verified
//
#include <hip/hip_runtime.h>

typedef __attribute__((ext_vector_type(2))) float v2f;
typedef __attribute__((ext_vector_type(4))) float f4;
typedef __attribute__((ext_vector_type(8))) float v8f;

#define NNODES 30000
#define NEDGES 480000
#define PBASIS 8

// ---------------------------------------------------------------- relu ----
__global__ void relu_kernel(const float* __restrict__ in, float* __restrict__ out, int n) {
    int i = blockIdx.x * blockDim.x + threadIdx.x;
    if (i < n) out[i] = fmaxf(in[i], 0.0f);
}

// ---------------------------------------------- node linear + bias init ---
// out[n*ldOut + colOff + co] = (fb?fb[co]:0) + (cb?cb[co]:0)
//                              + sum_ci act[n*ldIn+ci]*fW[ci*Cout+co]
//                              + (res ? res[n*ldOut+co] : 0)
// Pre-initializes the conv accumulation target so edge atomics add on top.
__global__ void node_linear_kernel(const float* __restrict__ act, int ldIn, int Cin,
                                   int Cout, int colOff, int ldOut,
                                   const float* __restrict__ fW,
                                   const float* __restrict__ fb,
                                   const float* __restrict__ cb,
                                   const float* __restrict__ res,
                                   float* __restrict__ out, int nNodes) {
    int idx = blockIdx.x * blockDim.x + threadIdx.x;
    if (idx >= nNodes * Cout) return;
    int co = idx % Cout;
    int n  = idx / Cout;
    float acc = 0.0f;
    if (fb) acc += fb[co];
    if (cb) acc += cb[co];
    for (int ci = 0; ci < Cin; ++ci)
        acc += act[n * ldIn + ci] * fW[ci * Cout + co];
    if (res) acc += res[n * ldOut + co];
    out[n * ldOut + colOff + co] = acc;
}

// --------------------------------------------------- fused edge WMMA -----
// msg[16xCout] = (phi ⊗ xj)[16 x P*CIN] @ W_flat[P*CIN x Cout], via
// V_WMMA_F32_16X16X4_F32 with W resident in LDS; f32 atomic scatter to fi.
// fi_s / fj_s hold PRE-SCALED int offsets (node*ld) so the hot gather and
// the atomic scatter need only 32-bit adds + one lshl_add_u64 each.
template <int CIN, int COUT>
__global__ __launch_bounds__(256) void edge_conv_kernel(
    const float* __restrict__ act, int ldIn,
    const int* __restrict__ fi, const int* __restrict__ fj,
    const float* __restrict__ dist,
    const float* __restrict__ cW,          // [P*CIN, COUT] row-major
    float* __restrict__ out, int ldOut, int colOff,
    int numTiles)
{
    constexpr int K      = PBASIS * CIN;              // 32 or 512
    constexpr int CINSH  = (CIN == 4) ? 2 : 6;        // log2(CIN)
    constexpr int COUTP  = (COUT < 16) ? 16 : COUT;   // pad to WMMA N tile
    constexpr int NT     = COUTP / 16;                // N tiles
    constexpr int LDW    = 72;                        // W LDS stride (bank-spread)
    constexpr int XJS    = 68;                        // xj LDS stride (bank-spread, 16B-mult)
    constexpr int WFLOATS = 16 * XJS + 16 * PBASIS + 32;

    extern __shared__ float smem[];
    float* W_lds = smem;                              // K * LDW floats

    const int tid  = threadIdx.x;
    const int wave = tid >> 5;
    const int lane = tid & 31;
    const int row  = lane & 15;                       // M (edge) / N (col) index
    const int half = lane >> 4;

    // Cooperative W_flat -> LDS, zero-padded to COUTP columns.
    for (int i = tid; i < K * COUTP; i += blockDim.x) {
        int k = i / COUTP, c = i % COUTP;
        W_lds[k * LDW + c] = (c < COUT) ? cW[k * COUT + c] : 0.0f;
    }
    __syncthreads();

    float* wbase = smem + K * LDW + wave * WFLOATS;
    float* xj_s  = wbase;                             // 16 x XJS
    float* phi_s = wbase + 16 * XJS;                  // 16 x 8
    int*   fi_s  = (int*)(wbase + 16 * XJS + 16 * PBASIS);  // pre-scaled fi*ldOut
    int*   fj_s  = fi_s + 16;                               // pre-scaled fj*ldIn

    const int wavesTotal = gridDim.x * (blockDim.x >> 5);
    const int gwave = blockIdx.x * (blockDim.x >> 5) + wave;

    for (int tile = gwave; tile < numTiles; tile += wavesTotal) {
        const int e0 = tile * 16;
        __builtin_amdgcn_wave_barrier();              // protect prev-iter LDS reads
        if (lane < 16) {
            int e = e0 + lane;
            int vfi = fi[e], vfj = fj[e];
            fi_s[lane] = vfi * ldOut;                 // pre-scaled scatter offset
            fj_s[lane] = vfj * ldIn;                  // pre-scaled gather offset
            float d = fminf(1.0f, fmaxf(-1.0f, dist[e]));
            float live = (vfi != vfj) ? 1.0f : 0.0f;  // centerIgnore folds into phi
#pragma unroll
            for (int p = 0; p < PBASIS; ++p) {
                float c  = -1.0f + (2.0f / 7.0f) * (float)p;
                float ph = fmaxf(0.0f, 1.0f - fabsf(d - c) * 3.5f);
                phi_s[lane * PBASIS + p] = ph * live;
            }
        }
        __builtin_amdgcn_wave_barrier();
        // Gather 16 source rows (L2-resident node table), float4 wide.
        constexpr int F4R = CIN / 4;
#pragma unroll
        for (int f = lane; f < 16 * F4R; f += 32) {
            int r = f / F4R, c4 = f % F4R;
            f4 v = *(const f4*)(act + fj_s[r] + c4 * 4);
            *(f4*)(xj_s + r * XJS + c4 * 4) = v;
        }
        __builtin_amdgcn_wave_barrier();

        v8f acc[NT] = {};
        const int ka0 = 2 * half;
#pragma unroll 2
        for (int kk = 0; kk < K; kk += 4) {
            const int ka = kk + ka0;
            // A tile 16x4: lane holds Z[row, ka] and Z[row, ka+1]
            float ax = phi_s[row * PBASIS + (ka >> CINSH)] *
                       xj_s[row * XJS + (ka & (CIN - 1))];
            float ay = phi_s[row * PBASIS + ((ka + 1) >> CINSH)] *
                       xj_s[row * XJS + ((ka + 1) & (CIN - 1))];
            v2f a; a.x = ax; a.y = ay;
#pragma unroll
            for (int t = 0; t < NT; ++t) {
                // B tile 4x16: lane holds W[ka, col] and W[ka+1, col]
                v2f b;
                b.x = W_lds[ka * LDW + t * 16 + row];
                b.y = W_lds[(ka + 1) * LDW + t * 16 + row];
                acc[t] = __builtin_amdgcn_wmma_f32_16x16x4_f32(
                    false, a, false, b, (short)0, acc[t], false, false);
            }
        }

        // D layout: VGPR v -> (M = v + 8*half, N = lane&15). Scatter to fi.
#pragma unroll
        for (int t = 0; t < NT; ++t) {
            int c = t * 16 + row;
            if (c < COUT) {
                float* colp = out + colOff + c;       // loop-invariant 64-bit base
#pragma unroll
                for (int v = 0; v < 8; ++v) {
                    int m = v + 8 * half;
                    unsafeAtomicAdd(colp + fi_s[m], acc[t][v]);  // 32-bit offset add
                }
            }
        }
    }
}

// -------------------------------------------------------------- launch ---
extern "C" void kernel_launch(void* const* d_in, const int* in_sizes, int n_in,
                              void* d_out, int out_size, void* d_ws, size_t ws_size,
                              hipStream_t stream) {
    const float* x    = (const float*)d_in[0];
    const int*   fi   = (const int*)d_in[1];
    const int*   fj   = (const int*)d_in[2];
    const float* dist = (const float*)d_in[3];
    const float* cW0 = (const float*)d_in[4];  const float* cb0 = (const float*)d_in[5];
    const float* fW0 = (const float*)d_in[6];  const float* fb0 = (const float*)d_in[7];
    const float* cW1 = (const float*)d_in[8];  const float* cb1 = (const float*)d_in[9];
    const float* fW1 = (const float*)d_in[10]; const float* fb1 = (const float*)d_in[11];
    const float* cW2 = (const float*)d_in[12]; const float* cb2 = (const float*)d_in[13];
    const float* fW2 = (const float*)d_in[14]; const float* fb2 = (const float*)d_in[15];
    const float* cW3 = (const float*)d_in[16]; const float* cb3 = (const float*)d_in[17];
    const float* fW3 = (const float*)d_in[18]; const float* fb3 = (const float*)d_in[19];
    float* outp = (float*)d_out;

    float* bufA   = (float*)d_ws;                     // [N,64]
    float* bufB   = bufA + (size_t)NNODES * 64;       // [N,64]
    float* bufAct = bufB + (size_t)NNODES * 64;       // [N,64]

    const int numTiles = NEDGES / 16;                 // 30000
    const dim3 blk(256);
    const int edgeBlocks = 640;                       // 5120 waves grid-striding tiles
    const size_t shBig   = (size_t)(512 * 72 + 8 * (16 * 68 + 16 * 8 + 32)) * 4;
    const size_t shSmall = (size_t)(32 * 72 + 8 * (16 * 68 + 16 * 8 + 32)) * 4;
    auto g1 = [](int n) { return dim3((n + 255) / 256); };

    // ---- layer 0: lin0 -> cols [0,32), conv0 (+cb0) -> cols [32,64)
    node_linear_kernel<<<g1(NNODES * 32), blk, 0, stream>>>(
        x, 4, 4, 32, 0, 64, fW0, fb0, nullptr, nullptr, bufA, NNODES);
    node_linear_kernel<<<g1(NNODES * 32), blk, 0, stream>>>(
        nullptr, 0, 0, 32, 32, 64, nullptr, cb0, nullptr, nullptr, bufA, NNODES);
    edge_conv_kernel<4, 32><<<edgeBlocks, blk, shSmall, stream>>>(
        x, 4, fi, fj, dist, cW0, bufA, 64, 32, numTiles);

    // ---- layer 1: 64 -> 64
    relu_kernel<<<g1(NNODES * 64), blk, 0, stream>>>(bufA, bufAct, NNODES * 64);
    node_linear_kernel<<<g1(NNODES * 64), blk, 0, stream>>>(
        bufAct, 64, 64, 64, 0, 64, fW1, fb1, cb1, nullptr, bufB, NNODES);
    edge_conv_kernel<64, 64><<<edgeBlocks, blk, shBig, stream>>>(
        bufAct, 64, fi, fj, dist, cW1, bufB, 64, 0, numTiles);

    // ---- layer 2: 64 -> 64 + residual(ans)
    relu_kernel<<<g1(NNODES * 64), blk, 0, stream>>>(bufB, bufAct, NNODES * 64);
    node_linear_kernel<<<g1(NNODES * 64), blk, 0, stream>>>(
        bufAct, 64, 64, 64, 0, 64, fW2, fb2, cb2, bufB, bufA, NNODES);
    edge_conv_kernel<64, 64><<<edgeBlocks, blk, shBig, stream>>>(
        bufAct, 64, fi, fj, dist, cW2, bufA, 64, 0, numTiles);

    // ---- layer 3: 64 -> 2, write d_out
    relu_kernel<<<g1(NNODES * 64), blk, 0, stream>>>(bufA, bufAct, NNODES * 64);
    node_linear_kernel<<<g1(NNODES * 2), blk, 0, stream>>>(
        bufAct, 64, 64, 2, 0, 2, fW3, fb3, cb3, nullptr, outp, NNODES);
    edge_conv_kernel<64, 2><<<edgeBlocks, blk, shBig, stream>>>(
        bufAct, 64, fi, fj, dist, cW3, outp, 2, 0, numTiles);
}